// RGCN_57578331570491
// MI455X (gfx1250) — compile-verified
//
#include <hip/hip_runtime.h>
#include <hip/hip_bf16.h>
#include <math.h>

typedef float v2f __attribute__((ext_vector_type(2)));
typedef float v8f __attribute__((ext_vector_type(8)));

#define DIN 128

// ---------------- per-column sum / sumsq (for BatchNorm over node dim) -------
__global__ __launch_bounds__(256)
void colstats_kernel(const float* __restrict__ X, float* __restrict__ stats,
                     long total, int D) {
    long stride = (long)gridDim.x * blockDim.x;
    long i0 = (long)blockIdx.x * blockDim.x + threadIdx.x;
    // stride is a multiple of D (D = 64 or 128, blockDim = 256) -> fixed column
    int col = (int)(i0 % (long)D);
    float s = 0.f, sq = 0.f;
    for (long i = i0; i < total; i += stride) {
        float v = X[i];
        s += v; sq += v * v;
    }
    atomicAdd(&stats[col], s);
    atomicAdd(&stats[D + col], sq);
}

__global__ void bn_finalize_kernel(const float* __restrict__ stats,
                                   const float* __restrict__ gamma,
                                   const float* __restrict__ beta,
                                   float* __restrict__ scale, float* __restrict__ shift,
                                   float invN, int D) {
    int c = threadIdx.x;
    if (c < D) {
        float mean = stats[c] * invN;
        float var  = stats[D + c] * invN - mean * mean;
        float r    = rsqrtf(var + 1e-5f);
        float sc   = gamma[c] * r;
        scale[c] = sc;
        shift[c] = beta[c] - mean * sc;
    }
}

// ---------------- in-degree (float so the divide matches the reference) -----
__global__ __launch_bounds__(256)
void degree_kernel(const int* __restrict__ dst, float* __restrict__ deg, int E) {
    int stride = gridDim.x * blockDim.x;
    for (int i = blockIdx.x * blockDim.x + threadIdx.x; i < E; i += stride)
        atomicAdd(&deg[dst[i]], 1.0f);
}

// deg[i] <- 1 / (deg[i] + 1): one IEEE divide per (graph,node) instead of
// 3 divides per combined feature element.
__global__ __launch_bounds__(256)
void invdeg_kernel(float* __restrict__ deg, long n) {
    long i = (long)blockIdx.x * blockDim.x + threadIdx.x;
    if (i < n) deg[i] = 1.0f / (deg[i] + 1.0f);
}

// ---------------- fused BN-apply + 3-weight GEMM via fp32 WMMA --------------
// Y[g][N][DOUT] = (scale*X+shift) @ Wg    (K = 128 fixed)
template <int DOUT>
__global__ __launch_bounds__(256)
void gemm_bn3_kernel(const float* __restrict__ X,
                     const float* __restrict__ scale, const float* __restrict__ shift,
                     const float* __restrict__ W0, const float* __restrict__ W1,
                     const float* __restrict__ W2,
                     float* __restrict__ Y, long N) {
    __shared__ float xs[16][132];                 // pad 128->132: conflict-free A loads
    const int tid = threadIdx.x;
    const long m0 = (long)blockIdx.x * 16;

    for (int i = tid; i < 16 * DIN; i += 256) {   // stage BN(x) tile in LDS
        int r = i >> 7, c = i & 127;
        xs[r][c] = X[(m0 + r) * DIN + c] * scale[c] + shift[c];
    }
    __syncthreads();

    const int wave = tid >> 5, lane = tid & 31;
    const int half = lane >> 4, l = lane & 15;    // A: lanes16-31 hold K+2..K+3
    const int tilesPerGraph = DOUT / 16;
    const int ntiles = 3 * tilesPerGraph;

    for (int t = wave; t < ntiles; t += 8) {      // wave-uniform -> EXEC all 1s
        const int g  = t / tilesPerGraph;
        const int n0 = (t % tilesPerGraph) * 16;
        const float* __restrict__ Wg = (g == 0) ? W0 : ((g == 1) ? W1 : W2);
        v8f acc = {};
        #pragma unroll 4
        for (int kk = 0; kk < 32; ++kk) {         // K=128 in steps of 4
            const int k = kk * 4 + half * 2;
            v2f a, b;
            a.x = xs[l][k];                       // A[m=l][k], A[m=l][k+1]
            a.y = xs[l][k + 1];
            b.x = Wg[k * DOUT + n0 + l];          // B[k][n=l], B[k+1][n=l]
            b.y = Wg[(k + 1) * DOUT + n0 + l];
            acc = __builtin_amdgcn_wmma_f32_16x16x4_f32(
                      false, a, false, b, (short)0, acc, false, false);
        }
        float* __restrict__ Yg = Y + ((long)g * N + m0) * DOUT;
        #pragma unroll
        for (int r = 0; r < 8; ++r) {             // D: VGPR r -> M=r (+8 for hi half)
            const int row = r + half * 8;
            Yg[(long)row * DOUT + n0 + l] = acc[r];
        }
    }
}

// ---------------- edge scatter-add of projected features --------------------
// D=128: 1 edge/wave (32 lanes x float4); D=64: 2 edges/wave (16 lanes each)
template <int D>
__global__ __launch_bounds__(256)
void scatter_add_kernel(const float* __restrict__ Y, const int* __restrict__ src,
                        const int* __restrict__ dst, float* __restrict__ acc,
                        int E) {
    constexpr int LPE = D / 4;        // lanes per edge (float4 granules)
    constexpr int EPW = 32 / LPE;     // edges per wave
    const int lane = threadIdx.x & 31;
    const int sub  = lane / LPE;      // edge slot within the wave
    const int cl   = lane % LPE;      // float4 index within the row
    long w  = (((long)blockIdx.x * blockDim.x) + threadIdx.x) >> 5;
    long nw = ((long)gridDim.x * blockDim.x) >> 5;
    for (long e = w * EPW + sub; e < (long)E; e += nw * EPW) {
        const long s = src[e], d = dst[e];
        float4 v = ((const float4*)(Y + s * D))[cl];
        float* __restrict__ ad = acc + d * D + 4 * cl;
        atomicAdd(ad + 0, v.x);
        atomicAdd(ad + 1, v.y);
        atomicAdd(ad + 2, v.z);
        atomicAdd(ad + 3, v.w);
    }
}

// ---------------- combine 3 relations: (+self)*invdeg + bias (+ReLU) --------
// also accumulates next-BN column stats in the same pass
template <int D, bool RELU>
__global__ __launch_bounds__(256)
void combine_kernel(const float* __restrict__ acc, const float* __restrict__ y,
                    const float* __restrict__ invdeg,
                    const float* __restrict__ b0, const float* __restrict__ b1,
                    const float* __restrict__ b2,
                    float* __restrict__ H, float* __restrict__ stats, long N) {
    const long total = N * D;
    const long stride = (long)gridDim.x * blockDim.x;
    const long i0 = (long)blockIdx.x * blockDim.x + threadIdx.x;
    const int col = (int)(i0 % (long)D);          // stride multiple of D
    const float bsum = b0[col] + b1[col] + b2[col];
    float s = 0.f, sq = 0.f;
    for (long i = i0; i < total; i += stride) {
        const long row = i / D;
        float v = bsum;
        #pragma unroll
        for (int g = 0; g < 3; ++g) {
            const float idg = invdeg[(long)g * N + row];
            v += (acc[(long)g * total + i] + y[(long)g * total + i]) * idg;
        }
        if (RELU) v = fmaxf(v, 0.0f);
        H[i] = v;
        s += v; sq += v * v;
    }
    atomicAdd(&stats[col], s);
    atomicAdd(&stats[D + col], sq);
}

// ---------------- BN3 + sigmoid + row min-max + row L2 normalize ------------
__global__ __launch_bounds__(256)
void final_rows_kernel(const float* __restrict__ Z, const float* __restrict__ scale,
                       const float* __restrict__ shift, float* __restrict__ out, long N) {
    const int lane = threadIdx.x & 31;
    const long row = (long)blockIdx.x * 8 + (threadIdx.x >> 5);
    if (row >= N) return;
    const float* __restrict__ zr = Z + row * 64;
    float v0 = zr[lane]      * scale[lane]      + shift[lane];
    float v1 = zr[lane + 32] * scale[lane + 32] + shift[lane + 32];
    v0 = 1.0f / (1.0f + expf(-v0));
    v1 = 1.0f / (1.0f + expf(-v1));
    float mx = fmaxf(v0, v1), mn = fminf(v0, v1);
    for (int off = 16; off; off >>= 1) {
        mx = fmaxf(mx, __shfl_xor(mx, off, 32));
        mn = fminf(mn, __shfl_xor(mn, off, 32));
    }
    const float inv = 1.0f / (mx - mn);
    v0 = (v0 - mn) * inv;
    v1 = (v1 - mn) * inv;
    float ss = v0 * v0 + v1 * v1;
    for (int off = 16; off; off >>= 1) ss += __shfl_xor(ss, off, 32);
    const float nrm = fmaxf(sqrtf(ss), 1e-12f);
    out[row * 64 + lane]      = v0 / nrm;
    out[row * 64 + lane + 32] = v1 / nrm;
}

extern "C" void kernel_launch(void* const* d_in, const int* in_sizes, int n_in,
                              void* d_out, int out_size, void* d_ws, size_t ws_size,
                              hipStream_t stream) {
    const float* x = (const float*)d_in[0];
    const int* srcp[3] = { (const int*)d_in[1], (const int*)d_in[3], (const int*)d_in[5] };
    const int* dstp[3] = { (const int*)d_in[2], (const int*)d_in[4], (const int*)d_in[6] };
    const float* W1g[3] = { (const float*)d_in[7],  (const float*)d_in[9],  (const float*)d_in[11] };
    const float* b1g[3] = { (const float*)d_in[8],  (const float*)d_in[10], (const float*)d_in[12] };
    const float* W2g[3] = { (const float*)d_in[13], (const float*)d_in[15], (const float*)d_in[17] };
    const float* b2g[3] = { (const float*)d_in[14], (const float*)d_in[16], (const float*)d_in[18] };
    const float* g1  = (const float*)d_in[19]; const float* be1 = (const float*)d_in[20];
    const float* g2  = (const float*)d_in[21]; const float* be2 = (const float*)d_in[22];
    const float* g3  = (const float*)d_in[23]; const float* be3 = (const float*)d_in[24];

    const long N = in_sizes[0] / DIN;             // 50000 (multiple of 16)
    const int  E = in_sizes[1];                   // 800000

    // ---- workspace layout (floats); y2/acc2/zpre alias layer-1 regions -----
    float* ws = (float*)d_ws;
    float* y1    = ws;                            // 3*N*128
    float* acc1  = y1   + 3 * N * DIN;            // 3*N*128
    float* deg   = acc1 + 3 * N * DIN;            // 3*N (becomes 1/(deg+1))
    float* h     = deg  + 3 * N;                  // N*128
    float* small = h    + N * DIN;
    float* stats1 = small;        float* sc1 = stats1 + 256; float* sh1 = sc1 + 128;
    float* stats2 = sh1 + 128;    float* sc2 = stats2 + 256; float* sh2 = sc2 + 128;
    float* stats3 = sh2 + 128;    float* sc3 = stats3 + 128; float* sh3 = sc3 + 64;
    float* y2   = y1;                             // 3*N*64  (y1 dead after combine1)
    float* acc2 = acc1;                           // 3*N*64  (acc1 dead after combine1)
    float* zpre = acc1 + 3 * N * 64;              // N*64    (disjoint from acc2)

    const float invN = 1.0f / (float)N;

    // ---- zero accumulators / stats ----
    hipMemsetAsync(acc1, 0, (size_t)3 * N * DIN * sizeof(float), stream);
    hipMemsetAsync(deg,  0, (size_t)3 * N * sizeof(float), stream);
    hipMemsetAsync(small, 0, (size_t)1312 * sizeof(float), stream);

    // ---- BN1 ----
    colstats_kernel<<<512, 256, 0, stream>>>(x, stats1, N * DIN, DIN);
    bn_finalize_kernel<<<1, 128, 0, stream>>>(stats1, g1, be1, sc1, sh1, invN, DIN);

    // ---- degrees -> reciprocal (1/(deg+1)) ----
    for (int g = 0; g < 3; ++g)
        degree_kernel<<<1024, 256, 0, stream>>>(dstp[g], deg + (long)g * N, E);
    invdeg_kernel<<<(int)((3 * N + 255) / 256), 256, 0, stream>>>(deg, 3 * N);

    // ---- layer 1: y1 = BN1(x) @ W_g1 ; scatter ; combine+ReLU ----
    gemm_bn3_kernel<DIN><<<(int)(N / 16), 256, 0, stream>>>(
        x, sc1, sh1, W1g[0], W1g[1], W1g[2], y1, N);
    for (int g = 0; g < 3; ++g)
        scatter_add_kernel<DIN><<<(E + 7) / 8, 256, 0, stream>>>(
            y1 + (long)g * N * DIN, srcp[g], dstp[g], acc1 + (long)g * N * DIN, E);
    combine_kernel<DIN, true><<<512, 256, 0, stream>>>(
        acc1, y1, deg, b1g[0], b1g[1], b1g[2], h, stats2, N);
    bn_finalize_kernel<<<1, 128, 0, stream>>>(stats2, g2, be2, sc2, sh2, invN, DIN);

    // ---- layer 2: y2 = BN2(h) @ W_g2 ; scatter ; combine ----
    hipMemsetAsync(acc2, 0, (size_t)3 * N * 64 * sizeof(float), stream);
    gemm_bn3_kernel<64><<<(int)(N / 16), 256, 0, stream>>>(
        h, sc2, sh2, W2g[0], W2g[1], W2g[2], y2, N);
    for (int g = 0; g < 3; ++g)
        scatter_add_kernel<64><<<(E / 2 + 7) / 8, 256, 0, stream>>>(
            y2 + (long)g * N * 64, srcp[g], dstp[g], acc2 + (long)g * N * 64, E);
    combine_kernel<64, false><<<512, 256, 0, stream>>>(
        acc2, y2, deg, b2g[0], b2g[1], b2g[2], zpre, stats3, N);
    bn_finalize_kernel<<<1, 64, 0, stream>>>(stats3, g3, be3, sc3, sh3, invN, 64);

    // ---- epilogue: BN3 + sigmoid + minmax + L2 ----
    final_rows_kernel<<<(int)((N + 7) / 8), 256, 0, stream>>>(
        zpre, sc3, sh3, (float*)d_out, N);
}